// MDescAug_31396210934413
// MI455X (gfx1250) — compile-verified
//
#include <hip/hip_runtime.h>

typedef float v2f __attribute__((ext_vector_type(2)));
typedef float v8f __attribute__((ext_vector_type(8)));

#define MTOP 30
#define KNB  20
#define DDIM 512
#define DP   516            // padded LDS row stride (floats) to avoid bank conflicts
#define BETA 0.15f
#define NEG_INF (-3.0e38f)

// Manual LDS arena layout (bytes) — Xt at 0 so sim-loop DS offsets fit the
// 16-bit immediate field (row base <= 63984, k-offset <= 2048).
#define XT_OFF   0
#define XT_BYTES (32 * DP * 4)                 // 66048
#define WD_OFF   (XT_OFF + XT_BYTES)           // 66048
#define WD_BYTES (32 * 36 * 4)                 // 4608
#define SIM_OFF  (WD_OFF + WD_BYTES)           // 70656
#define SIM_BYTES (32 * 33 * 4)                // 4224
#define XD_OFF   (SIM_OFF + SIM_BYTES)         // 74880
#define XD_BYTES (32 * DP * 4)                 // 66048
#define QS_OFF   (XD_OFF + XD_BYTES)           // 140928
#define QS_BYTES (DDIM * 4)                    // 2048
#define IW_OFF   (QS_OFF + QS_BYTES)           // invw
#define SC_OFF   (IW_OFF + 128)                // sc
#define IDS_OFF  (SC_OFF + 128)                // ids
#define ORD_OFF  (IDS_OFF + 128)               // ord
#define SMEM_BYTES (ORD_OFF + 128)             // 143488

__global__ __launch_bounds__(128) void mdesc_aug_kernel(
    const float* __restrict__ X,
    const float* __restrict__ Q,
    const int*   __restrict__ ranks,   // [R, B] row-major; we use rows 0..29
    float* __restrict__ out,
    int B)
{
  __shared__ __align__(16) unsigned char smem[SMEM_BYTES];
  float (* const Xt)[DP] = (float (*)[DP])(smem + XT_OFF);   // rows 30,31 zero
  float (* const Wd)[36] = (float (*)[36])(smem + WD_OFF);   // dense weights
  float (* const sim)[33] = (float (*)[33])(smem + SIM_OFF);
  float (* const XD)[DP] = (float (*)[DP])(smem + XD_OFF);   // x_dba
  float* const Qs   = (float*)(smem + QS_OFF);
  float* const invw = (float*)(smem + IW_OFF);
  float* const sc   = (float*)(smem + SC_OFF);
  int*   const ids  = (int*)(smem + IDS_OFF);
  int*   const ord  = (int*)(smem + ORD_OFF);

  const int b    = blockIdx.x;
  const int tid  = threadIdx.x;
  const int wave = tid >> 5;
  const int lane = tid & 31;

  // ---- candidate ids: rt[m] = ranks[m, b]
  if (tid < MTOP) ids[tid] = ranks[(size_t)tid * (size_t)B + (size_t)b];
  for (int i = tid; i < 32 * 36; i += 128) ((float*)Wd)[i] = 0.0f;
  __syncthreads();

  // ---- gather Xt rows from HBM into LDS (float4), zero pad rows 30,31
  for (int i = tid; i < MTOP * (DDIM / 4); i += 128) {
    const int m = i >> 7, c = i & 127;
    const float4* src = (const float4*)(X + (size_t)ids[m] * DDIM);
    ((float4*)&Xt[m][0])[c] = src[c];
  }
  for (int i = tid; i < 2 * (DDIM / 4); i += 128) {
    const int m = 30 + (i >> 7);
    ((float4*)&Xt[m][0])[i & 127] = make_float4(0.f, 0.f, 0.f, 0.f);
  }
  for (int i = tid; i < DDIM / 4; i += 128)
    ((float4*)Qs)[i] = ((const float4*)(Q + (size_t)b * DDIM))[i];
  __syncthreads();

  // ---- sim = Xt * Xt^T via V_WMMA_F32_16X16X4_F32, one 16x16 tile per wave.
  // A-frag (16x4 f32): lanes 0-15 hold K=k,k+1 ; lanes 16-31 hold K=k+2,k+3.
  // B-frag (4x16) mirrors A since B[k][n] = Xt[n][k].
  // Two independent accumulator chains so successive WMMAs can overlap.
  {
    const int I  = wave >> 1, J = wave & 1;
    const int r  = 16 * I + (lane & 15);
    const int cc = 16 * J + (lane & 15);
    const int kh = 2 * (lane >> 4);
    v8f acc0 = {}, acc1 = {};
    for (int k = 0; k < DDIM; k += 8) {
      v2f a0 = *(const v2f*)&Xt[r][k + kh];
      v2f b0 = *(const v2f*)&Xt[cc][k + kh];
      v2f a1 = *(const v2f*)&Xt[r][k + 4 + kh];
      v2f b1 = *(const v2f*)&Xt[cc][k + 4 + kh];
      acc0 = __builtin_amdgcn_wmma_f32_16x16x4_f32(false, a0, false, b0,
                                                   (short)0, acc0, false, false);
      acc1 = __builtin_amdgcn_wmma_f32_16x16x4_f32(false, a1, false, b1,
                                                   (short)0, acc1, false, false);
    }
    for (int v = 0; v < 8; ++v)
      sim[16 * I + v + 8 * (lane >> 4)][16 * J + (lane & 15)] = acc0[v] + acc1[v];
  }
  __syncthreads();

  // ---- top-K_NB selection per row (strict > scan == lowest-index tie-break,
  // matching jax.lax.top_k); scatter weights into dense Wd.
  if (tid < MTOP) {
    float vbuf[MTOP];
    for (int j = 0; j < MTOP; ++j) vbuf[j] = sim[tid][j];
    float wsum = 0.f;
    for (int k = 0; k < KNB; ++k) {
      float best = NEG_INF; int bj = 0;
      for (int j = 0; j < MTOP; ++j)
        if (vbuf[j] > best) { best = vbuf[j]; bj = j; }
      const float w = (k == 0) ? 1.0f : BETA * best;
      Wd[tid][bj] += w;
      wsum += w;
      vbuf[bj] = NEG_INF;
    }
    invw[tid] = 1.0f / wsum;
  } else if (tid < 32) {
    invw[tid] = 0.f;
  }
  __syncthreads();

  const size_t oM = (size_t)B * MTOP;   // section stride in out[]

  // ---- x_dba = (Wd @ Xt) * invw : 32x32 x 32x512 GEMM, WMMA f32 16x16x4.
  // 64 output tiles (2 Mtiles x 32 Ntiles) over 4 waves; two N-tiles at a time
  // (same Mt => shared A-fragment, two independent accumulator chains).
  {
    const int kh = 2 * (lane >> 4);
    for (int t = wave; t < 64; t += 8) {
      const int Mt   = t & 1;
      const int Nt0  = t >> 1;
      const int Nt1  = (t + 4) >> 1;
      const int ar   = 16 * Mt + (lane & 15);
      const int col0 = 16 * Nt0 + (lane & 15);
      const int col1 = 16 * Nt1 + (lane & 15);
      v8f acc0 = {}, acc1 = {};
      for (int k = 0; k < 32; k += 4) {
        v2f a;
        a.x = Wd[ar][k + kh];
        a.y = Wd[ar][k + kh + 1];
        v2f b0, b1;
        b0.x = Xt[k + kh][col0];
        b0.y = Xt[k + kh + 1][col0];
        b1.x = Xt[k + kh][col1];
        b1.y = Xt[k + kh + 1][col1];
        acc0 = __builtin_amdgcn_wmma_f32_16x16x4_f32(false, a, false, b0,
                                                     (short)0, acc0, false, false);
        acc1 = __builtin_amdgcn_wmma_f32_16x16x4_f32(false, a, false, b1,
                                                     (short)0, acc1, false, false);
      }
      for (int v = 0; v < 8; ++v) {
        const int row = 16 * Mt + v + 8 * (lane >> 4);
        const float s = invw[row];
        const float v0 = acc0[v] * s;
        const float v1 = acc1[v] * s;
        XD[row][col0] = v0;
        XD[row][col1] = v1;
        if (row < MTOP) {
          out[3 * oM + ((size_t)b * MTOP + row) * DDIM + col0] = v0;  // x_dba
          out[3 * oM + ((size_t)b * MTOP + row) * DDIM + col1] = v1;
        }
      }
    }
  }
  __syncthreads();

  // ---- scores[m] = Q . x_dba[m]  (wave-parallel rows, lane-split dot)
  for (int m = wave; m < MTOP; m += 4) {
    float s = 0.f;
    for (int d = lane; d < DDIM; d += 32) s += Qs[d] * XD[m][d];
    for (int off = 16; off > 0; off >>= 1) s += __shfl_xor(s, off, 32);
    if (lane == 0) sc[m] = s;
  }
  __syncthreads();

  // ---- stable argsort descending (rank by count), then outputs
  if (tid < MTOP) {
    const float si = sc[tid];
    int pos = 0;
    for (int j = 0; j < MTOP; ++j) {
      const float sj = sc[j];
      if (sj > si || (sj == si && j < tid)) ++pos;
    }
    ord[pos] = tid;
  }
  __syncthreads();

  if (tid < MTOP) {
    const int i = ord[tid];                       // element at sorted position tid
    const size_t base = (size_t)b * MTOP + tid;
    out[base]          = (float)ids[i];           // rerank = rt[order]
    out[oM + base]     = sc[tid];                 // scores (per-m, unsorted)
    out[2 * oM + base] = (float)i;                // order
  }
}

extern "C" void kernel_launch(void* const* d_in, const int* in_sizes, int n_in,
                              void* d_out, int out_size, void* d_ws, size_t ws_size,
                              hipStream_t stream) {
  const float* X     = (const float*)d_in[0];   // [N, 512] f32
  const float* Q     = (const float*)d_in[1];   // [B, 512] f32
  const int*   ranks = (const int*)d_in[2];     // [R, B] int32
  float*       out   = (float*)d_out;

  const int B = in_sizes[1] / DDIM;             // 4096

  mdesc_aug_kernel<<<B, 128, 0, stream>>>(X, Q, ranks, out, B);
}